// SepTemporalCrossAttention_446676599072
// MI455X (gfx1250) — compile-verified
//
#include <hip/hip_runtime.h>
#include <math.h>

// Problem constants (from reference)
#define B_    256
#define T_    10
#define N_    24
#define D_    512
#define H_    8
#define DH_   64
#define MAXREL 8
#define BT_   (B_ * T_)            // 2560
#define ROWS3 (B_ * T_ * N_)       // 61440
#define OUT_ELEMS ((size_t)B_ * T_ * N_ * D_)   // 31457280

typedef __attribute__((ext_vector_type(2))) float v2f;
typedef __attribute__((ext_vector_type(8))) float v8f;

// ---------------------------------------------------------------------------
// Positional encoding (computed in double to match numpy's float64 promotion)
// pe[t*D + d] = sin/cos( t / 10000^(2*floor(d/2)/D) )
// ---------------------------------------------------------------------------
__global__ void pe_kernel(float* __restrict__ pe) {
  int i = blockIdx.x * blockDim.x + threadIdx.x;
  if (i >= T_ * D_) return;
  int d = i % D_, t = i / D_;
  double e = 2.0 * (double)(d >> 1) / (double)D_;
  double ang = (double)t * pow(10000.0, -e);
  pe[i] = (float)((d & 1) ? cos(ang) : sin(ang));
}

// peW[n, t, m] = sum_d pe[t, d] * Wq[n, d, m]   (folds pos-enc into q bias)
__global__ void peW_kernel(const float* __restrict__ pe,
                           const float* __restrict__ Wq,
                           float* __restrict__ peW) {
  int i = blockIdx.x * blockDim.x + threadIdx.x;
  if (i >= N_ * T_ * D_) return;
  int m = i % D_;
  int t = (i / D_) % T_;
  int n = i / (D_ * T_);
  const float* w = Wq + (size_t)n * D_ * D_ + m;
  const float* p = pe + t * D_;
  float s = 0.f;
#pragma unroll 4
  for (int d = 0; d < D_; ++d) s += p[d] * w[(size_t)d * D_];
  peW[i] = s;
}

// ---------------------------------------------------------------------------
// Generic fp32 WMMA GEMM:  Y[r, n, m] = sum_k A[r, n, k] * W[n, k, m] + bias
// Each wave computes a 32x64 tile via V_WMMA_F32_16X16X4_F32.
// A fragment layout (16x4 f32, MxK): lane 0-15 = M, VGPR0/1 = K+{0,1},
//   lanes 16-31 carry K+{2,3}.  B (4x16, KxN) mirrors with lane = N column.
// C/D (16x16 f32): VGPR j -> row j (lanes 0-15) / row j+8 (lanes 16-31).
// ---------------------------------------------------------------------------
__global__ __launch_bounds__(256) void wmma_gemm_kernel(
    const float* __restrict__ A, long lda, long aNStride,
    const float* __restrict__ Wbase, const float* __restrict__ bias,
    const float* __restrict__ peW,            // nullable: adds peW[n, r%T, m]
    float* __restrict__ Y, long ldy, long yNStride,
    int Mtiles32, int totalJobs)
{
  int job = blockIdx.x * (blockDim.x >> 5) + (threadIdx.x >> 5);
  if (job >= totalJobs) return;
  const int NT = D_ / 64;                       // 8 column tiles of 64
  int n     = job / (Mtiles32 * NT);
  int rem   = job - n * (Mtiles32 * NT);
  int mtile = rem / NT;
  int ntile = rem - mtile * NT;

  int lane = threadIdx.x & 31;
  int lh   = lane >> 4;      // lane half: selects K+0/1 vs K+2/3
  int lm   = lane & 15;      // M row (A) / N column (B) within tile

  const float* aPtr0 = A + (size_t)(mtile * 32 + lm) * lda
                         + (size_t)n * aNStride + 2 * lh;
  const float* aPtr1 = aPtr0 + (size_t)16 * lda;
  const float* wB = Wbase + (size_t)n * D_ * D_
                          + (size_t)(2 * lh) * D_ + (ntile * 64 + lm);

  v8f acc[2][4] = {};

  for (int kk = 0; kk < D_; kk += 4) {
    v2f a0 = *(const v2f*)(aPtr0 + kk);
    v2f a1 = *(const v2f*)(aPtr1 + kk);
    const float* wp = wB + (size_t)kk * D_;
#pragma unroll
    for (int sub = 0; sub < 4; ++sub) {
      v2f b;
      b.x = wp[sub * 16];
      b.y = wp[sub * 16 + D_];
      acc[0][sub] = __builtin_amdgcn_wmma_f32_16x16x4_f32(
          false, a0, false, b, (short)0, acc[0][sub], false, false);
      acc[1][sub] = __builtin_amdgcn_wmma_f32_16x16x4_f32(
          false, a1, false, b, (short)0, acc[1][sub], false, false);
    }
  }

  const float* bN = bias + (size_t)n * D_;
#pragma unroll
  for (int mi = 0; mi < 2; ++mi) {
    int rBase = mtile * 32 + mi * 16 + 8 * lh;
#pragma unroll
    for (int sub = 0; sub < 4; ++sub) {
      int m = ntile * 64 + sub * 16 + lm;
      float bb = bN[m];
#pragma unroll
      for (int j = 0; j < 8; ++j) {
        int r = rBase + j;
        float val = acc[mi][sub][j] + bb;
        if (peW) val += peW[((size_t)n * T_ + (r % T_)) * D_ + m];
        Y[(size_t)r * ldy + (size_t)n * yNStride + m] = val;
      }
    }
  }
}

// ---------------------------------------------------------------------------
// Attention per (b, n, h): T=10, dh=64.  One 64-thread block per unit.
// logits[q,k] = q . (k + key_table[idx(q,k)]) / 8 + mask*-1e9 ; softmax ;
// out[q,d] = sum_t w[q,t] * (v[t,d] + val_table[idx(q,t)][d]).
// Writes attn-out IN PLACE over the q buffer (block owns its slice; all
// reads are staged to LDS before the barrier).  Also emits last_attn row.
// ---------------------------------------------------------------------------
__global__ __launch_bounds__(64) void attention_kernel(
    float* __restrict__ qbuf,            // in: q, out: attention output
    const float* __restrict__ kbuf,
    const float* __restrict__ vbuf,
    const float* __restrict__ mask,      // (T,T)
    const float* __restrict__ key_table, // (17, dh)
    const float* __restrict__ val_table, // (17, dh)
    float* __restrict__ last_attn)       // (B,N,H,T)
{
  int unit = blockIdx.x;                 // b*N*H + n*H + h
  int h = unit % H_;
  int n = (unit / H_) % N_;
  int b = unit / (H_ * N_);
  int tid = threadIdx.x;                 // 0..63 = d

  __shared__ float qs[T_][DH_], ks[T_][DH_], vs[T_][DH_];
  __shared__ float lg[T_][T_], wr[T_][T_];

  for (int t = 0; t < T_; ++t) {
    size_t base = (((size_t)(b * T_ + t) * N_) + n) * D_ + h * DH_ + tid;
    qs[t][tid] = qbuf[base];
    ks[t][tid] = kbuf[base];
    vs[t][tid] = vbuf[base];
  }
  __syncthreads();

  for (int p = tid; p < T_ * T_; p += 64) {
    int qi = p / T_, ki = p - qi * T_;
    int rel = ki - qi;
    rel = rel < -MAXREL ? -MAXREL : (rel > MAXREL ? MAXREL : rel);
    const float* rk = key_table + (rel + MAXREL) * DH_;
    float s = 0.f;
#pragma unroll
    for (int d = 0; d < DH_; ++d) s += qs[qi][d] * (ks[ki][d] + rk[d]);
    lg[qi][ki] = s * 0.125f + mask[qi * T_ + ki] * -1.0e9f;
  }
  __syncthreads();

  if (tid < T_) {
    float mx = lg[tid][0];
    for (int k = 1; k < T_; ++k) mx = fmaxf(mx, lg[tid][k]);
    float sum = 0.f;
    for (int k = 0; k < T_; ++k) {
      float e = __expf(lg[tid][k] - mx);
      wr[tid][k] = e;
      sum += e;
    }
    float inv = 1.f / sum;
    for (int k = 0; k < T_; ++k) wr[tid][k] *= inv;
  }
  __syncthreads();

  for (int qi = 0; qi < T_; ++qi) {
    float acc = 0.f;
#pragma unroll
    for (int t = 0; t < T_; ++t) {
      int rel = t - qi;
      rel = rel < -MAXREL ? -MAXREL : (rel > MAXREL ? MAXREL : rel);
      acc += wr[qi][t] * (vs[t][tid] + val_table[(rel + MAXREL) * DH_ + tid]);
    }
    qbuf[(((size_t)(b * T_ + qi) * N_) + n) * D_ + h * DH_ + tid] = acc;
  }

  if (tid < T_)
    last_attn[(((size_t)(b * N_ + n) * H_) + h) * T_ + tid] = wr[T_ - 1][tid];
}

// ---------------------------------------------------------------------------
extern "C" void kernel_launch(void* const* d_in, const int* in_sizes, int n_in,
                              void* d_out, int out_size, void* d_ws, size_t ws_size,
                              hipStream_t stream) {
  const float* eps_v     = (const float*)d_in[0];
  const float* hist      = (const float*)d_in[1];
  const float* mask      = (const float*)d_in[2];
  const float* Wq        = (const float*)d_in[3];
  const float* bq        = (const float*)d_in[4];
  const float* Wk        = (const float*)d_in[5];
  const float* bk        = (const float*)d_in[6];
  const float* Wv        = (const float*)d_in[7];
  const float* bv        = (const float*)d_in[8];
  const float* Wo        = (const float*)d_in[9];
  const float* bo        = (const float*)d_in[10];
  const float* key_table = (const float*)d_in[11];
  const float* val_table = (const float*)d_in[12];

  float* out       = (float*)d_out;
  float* last_attn = out + OUT_ELEMS;

  // Workspace layout (floats): pe | peW | q(->attn_out in place) | k | v
  float* ws  = (float*)d_ws;
  float* pe  = ws;                          // 5120
  float* peW = ws + 5120;                   // 122880  (ends 128000)
  float* qb  = ws + 131072;                 // 31457280
  float* kb  = qb + OUT_ELEMS;
  float* vb  = kb + OUT_ELEMS;

  // 1) positional encoding + folded pe @ Wq bias
  pe_kernel<<<(T_ * D_ + 255) / 256, 256, 0, stream>>>(pe);
  peW_kernel<<<(N_ * T_ * D_ + 255) / 256, 256, 0, stream>>>(pe, Wq, peW);

  // 2) q/k/v projections: per-n (2560 x 512) @ (512 x 512) GEMMs
  const int mt1   = BT_ / 32;                       // 80
  const int jobs1 = N_ * mt1 * (D_ / 64);           // 15360
  const int blk1  = jobs1 / 8;                      // 1920 (8 waves/block)
  wmma_gemm_kernel<<<blk1, 256, 0, stream>>>(
      eps_v, (long)(N_ * D_), (long)D_, Wq, bq, peW,
      qb, (long)(N_ * D_), (long)D_, mt1, jobs1);
  wmma_gemm_kernel<<<blk1, 256, 0, stream>>>(
      hist, (long)(N_ * D_), (long)D_, Wk, bk, nullptr,
      kb, (long)(N_ * D_), (long)D_, mt1, jobs1);
  wmma_gemm_kernel<<<blk1, 256, 0, stream>>>(
      hist, (long)(N_ * D_), (long)D_, Wv, bv, nullptr,
      vb, (long)(N_ * D_), (long)D_, mt1, jobs1);

  // 3) attention (writes attn-out over qb, emits last_attn)
  attention_kernel<<<B_ * N_ * H_, 64, 0, stream>>>(
      qb, kb, vb, mask, key_table, val_table, last_attn);

  // 4) output projection: (61440 x 512) @ (512 x 512) -> d_out
  const int mt3   = ROWS3 / 32;                     // 1920
  const int jobs3 = mt3 * (D_ / 64);                // 15360
  const int blk3  = jobs3 / 8;                      // 1920
  wmma_gemm_kernel<<<blk3, 256, 0, stream>>>(
      qb, (long)D_, 0L, Wo, bo, nullptr,
      out, (long)D_, 0L, mt3, jobs3);
}